// TSBRNN_44246753083693
// MI455X (gfx1250) — compile-verified
//
#include <hip/hip_runtime.h>
#include <cstdint>

#define TT 2048                 // time steps per row
#define LANES 32                // wave32
#define CHUNK (TT / LANES)      // 64 steps per lane
#define WAVES_PER_BLOCK 4
#define BLOCK (WAVES_PER_BLOCK * LANES)
#define COPY_ITERS (TT * 4 / (LANES * 16))   // 16 x b128 issues per 8KB array

__global__ __launch_bounds__(BLOCK) void tsb_scan_kernel(
    const float* __restrict__ X,
    const int*   __restrict__ Xid,
    const float* __restrict__ alpha,
    const float* __restrict__ beta,
    float*       __restrict__ out)
{
    __shared__ float sx[WAVES_PER_BLOCK][TT];   // 32 KB
    __shared__ int   sid[WAVES_PER_BLOCK][TT];  // 32 KB

    const int lane = threadIdx.x & (LANES - 1);
    const int wave = threadIdx.x >> 5;
    const int row  = blockIdx.x * WAVES_PER_BLOCK + wave;

    const float* gx = X   + (size_t)row * TT;
    const int*   gi = Xid + (size_t)row * TT;
    float*       go = out + (size_t)row * TT;

    // Low 32 bits of a generic pointer into LDS == workgroup-relative LDS byte offset.
    uint32_t lds_x  = (uint32_t)(uintptr_t)&sx[wave][0];
    uint32_t lds_id = (uint32_t)(uintptr_t)&sid[wave][0];

    // ---- Stage this wave's X and X_id rows into LDS with CDNA5 async copies ----
    // 8KB per array, 32 lanes x 16B = 512B contiguous per issue.
    // Non-temporal: streamed once, keep L2 free for the alpha/beta gather tables.
#pragma unroll
    for (int it = 0; it < COPY_ITERS; ++it) {
        uint32_t off = (uint32_t)(it * LANES + lane) * 16u;   // byte offset in row
        uint32_t lx  = lds_x  + off;
        uint32_t li  = lds_id + off;
        asm volatile("global_load_async_to_lds_b128 %0, %1, %2 th:TH_LOAD_NT"
                     :: "v"(lx), "v"(off), "s"(gx) : "memory");
        asm volatile("global_load_async_to_lds_b128 %0, %1, %2 th:TH_LOAD_NT"
                     :: "v"(li), "v"(off), "s"(gi) : "memory");
    }
    asm volatile("s_wait_asynccnt 0x0" ::: "memory");

    const int base = lane * CHUNK;

    // ---- Pass 1: per-lane composition of affine maps over its chunk ----
    // Z' = az*Z + bz with az = 1 - nz*al, bz = nz*al*x
    // P' = ap*P + bp with ap = 1 - be,    bp = nz*be
    float Az = 1.f, Bz = 0.f, Ap = 1.f, Bp = 0.f;
#pragma unroll 8
    for (int j = 0; j < CHUNK; ++j) {
        float x  = sx[wave][base + j];
        int   id = sid[wave][base + j];
        float al = alpha[id];
        float be = beta[id];
        float nz = (x != 0.f) ? 1.f : 0.f;
        float na = nz * al;
        float az = 1.f - na;
        float bz = na * x;
        float ap = 1.f - be;
        float bp = nz * be;
        Bz = az * Bz + bz;  Az = az * Az;
        Bp = ap * Bp + bp;  Ap = ap * Ap;
    }

    // ---- Wave-level inclusive scan of affine pairs (Hillis–Steele, 5 steps) ----
#pragma unroll
    for (int d = 1; d < LANES; d <<= 1) {
        float pAz = __shfl_up(Az, d, LANES);
        float pBz = __shfl_up(Bz, d, LANES);
        float pAp = __shfl_up(Ap, d, LANES);
        float pBp = __shfl_up(Bp, d, LANES);
        if (lane >= d) {
            Bz = Az * pBz + Bz;  Az = Az * pAz;
            Bp = Ap * pBp + Bp;  Ap = Ap * pAp;
        }
    }
    // Exclusive: state entering my chunk = B-term of previous lane (global init is 0).
    float Z = __shfl_up(Bz, 1, LANES);
    float P = __shfl_up(Bp, 1, LANES);
    if (lane == 0) { Z = 0.f; P = 0.f; }

    // ---- Pass 2: replay chunk with correct initial state; overwrite x-slot with y ----
#pragma unroll 8
    for (int j = 0; j < CHUNK; ++j) {
        float x  = sx[wave][base + j];
        int   id = sid[wave][base + j];
        float al = alpha[id];
        float be = beta[id];
        float nz = (x != 0.f) ? 1.f : 0.f;
        float na = nz * al;
        Z = (1.f - na) * Z + na * x;
        P = (1.f - be) * P + nz * be;
        sx[wave][base + j] = P * Z;
    }

    // ---- Drain results LDS -> global with async stores (coalesced 512B/issue) ----
    // Non-temporal store: output is never re-read by the GPU.
    asm volatile("s_wait_dscnt 0x0" ::: "memory");
#pragma unroll
    for (int it = 0; it < COPY_ITERS; ++it) {
        uint32_t off = (uint32_t)(it * LANES + lane) * 16u;
        uint32_t lx  = lds_x + off;
        asm volatile("global_store_async_from_lds_b128 %0, %1, %2 th:TH_STORE_NT"
                     :: "v"(off), "v"(lx), "s"(go) : "memory");
    }
    asm volatile("s_wait_asynccnt 0x0" ::: "memory");
}

extern "C" void kernel_launch(void* const* d_in, const int* in_sizes, int n_in,
                              void* d_out, int out_size, void* d_ws, size_t ws_size,
                              hipStream_t stream) {
    (void)n_in; (void)out_size; (void)d_ws; (void)ws_size;
    const float* X     = (const float*)d_in[0];
    const int*   Xid   = (const int*)d_in[1];
    const float* alpha = (const float*)d_in[2];
    const float* beta  = (const float*)d_in[3];
    float*       out   = (float*)d_out;

    const int Brows = in_sizes[0] / TT;          // 4096
    dim3 grid(Brows / WAVES_PER_BLOCK);          // 1024 blocks x 4 waves = 1 wave/row
    tsb_scan_kernel<<<grid, BLOCK, 0, stream>>>(X, Xid, alpha, beta, out);
}